// SlidingWindowL2Pooling_64733747085817
// MI455X (gfx1250) — compile-verified
//
#include <hip/hip_runtime.h>

// Problem constants (match reference: x is (1024, 2048, 64) fp32, window 64)
#define BATCH   1024
#define SEQ     2048
#define DIM     64
#define WIN     64
#define NWIN    (SEQ - WIN)      // 1984 windows
#define NTHREADS 256
#define NWAVES   8
#define ROWPAD   68              // LDS row stride in floats (kills bank conflicts)
#define NTILES   (SEQ / 16)      // 128 16-row tiles per batch row

typedef __attribute__((ext_vector_type(2))) float v2f;
typedef __attribute__((ext_vector_type(8))) float v8f;

__device__ __forceinline__ float vsel8(v8f v, int i) {
    float r = v[0];
    r = (i == 1) ? v[1] : r;
    r = (i == 2) ? v[2] : r;
    r = (i == 3) ? v[3] : r;
    r = (i == 4) ? v[4] : r;
    r = (i == 5) ? v[5] : r;
    r = (i == 6) ? v[6] : r;
    r = (i == 7) ? v[7] : r;
    return r;
}

// Issue 8 async global->LDS copies (32 lanes x 16B each = one 4KB 16x64 tile,
// written into LDS with a 68-float row pitch). Tracked by ASYNCcnt; no VGPR
// data round-trip. lds_base is the LDS byte offset (low 32 bits of the flat
// address of a __shared__ object, per the ISA aperture rule).
__device__ __forceinline__ void issue_tile_async(const float* __restrict__ xb,
                                                 int g, unsigned lds_base,
                                                 int lane) {
    #pragma unroll
    for (int i = 0; i < 8; ++i) {
        int fidx = i * 32 + lane;           // 0..255 float4s
        int row  = fidx >> 4;               // 0..15
        int c4   = fidx & 15;               // 0..15
        unsigned goff  = (unsigned)(((g * 16 + row) * DIM + c4 * 4) * sizeof(float));
        unsigned laddr = lds_base + (unsigned)((row * ROWPAD + c4 * 4) * sizeof(float));
        asm volatile("global_load_async_to_lds_b128 %0, %1, %2"
                     :
                     : "v"(laddr), "v"(goff), "s"(xb)
                     : "memory");
    }
}

__launch_bounds__(NTHREADS, 2)
__global__ void SlidingWindowL2Pooling_kernel(const float* __restrict__ x,
                                              float* __restrict__ out) {
    __shared__ float stage[NWAVES][2][16 * ROWPAD]; // per-wave double buffer
    __shared__ float sq[SEQ];                       // per-row squared norms
    __shared__ float rv[NTHREADS];
    __shared__ int   ri[NTHREADS];

    const int b    = blockIdx.x;
    const int tid  = threadIdx.x;
    const int lane = tid & 31;
    const int wave = tid >> 5;

    const float* xb = x + (size_t)b * SEQ * DIM;

    // ---- Pass 1: squared row norms via WMMA Gram-diagonal ------------------
    // Each wave owns tiles g = wave, wave+8, ... (16 tiles), double-buffered:
    // async-copy tile g+8 into one buffer while WMMAs consume the other.
    issue_tile_async(xb, wave, (unsigned)(size_t)&stage[wave][0][0], lane);

    int ibuf = 0;
    for (int g = wave; g < NTILES; g += NWAVES) {
        if (g + NWAVES < NTILES) {
            issue_tile_async(xb, g + NWAVES,
                             (unsigned)(size_t)&stage[wave][ibuf ^ 1][0], lane);
            // Async loads complete in order: <=8 outstanding means the 8
            // copies for the CURRENT buffer have landed in LDS.
            asm volatile("s_wait_asynccnt 0x8" ::: "memory");
        } else {
            asm volatile("s_wait_asynccnt 0x0" ::: "memory");
        }

        // WMMA f32 16x16x4: A-layout for a 16x64 tile is identical to the
        // B(=A^T)-layout, so the same regs feed SRC0 and SRC1 -> C = A*A^T.
        // Direct __shared__ indexing => addrspace(3) => ds_load_b64.
        const int r16  = lane & 15;          // matrix row M
        const int kh   = (lane >> 4) << 1;   // K offset 0 (lanes 0-15) or 2 (16-31)
        const int soff = r16 * ROWPAD + kh;

        v8f acc = {};
        #pragma unroll
        for (int k0 = 0; k0 < 16; ++k0) {
            v2f a = *(const v2f*)&stage[wave][ibuf][soff + 4 * k0];
            acc = __builtin_amdgcn_wmma_f32_16x16x4_f32(
                /*neg_a=*/false, a, /*neg_b=*/false, a,
                /*c_mod=*/(short)0, acc, /*reuse_a=*/false, /*reuse_b=*/false);
        }

        // Gram diagonal: rows 0-7 live in lanes 0-7 (VGPR = lane),
        // rows 8-15 live in lanes 24-31 (VGPR = lane-24).
        if (lane < 8) {
            sq[g * 16 + lane] = vsel8(acc, lane);
        } else if (lane >= 24) {
            sq[g * 16 + lane - 16] = vsel8(acc, lane - 24);
        }
        ibuf ^= 1;
    }
    __syncthreads();

    // ---- Pass 2: sliding-window sums + argmax (first-index tiebreak) -------
    float bv = -__builtin_inff();
    int   bi = 0;
    {
        int j0 = tid * 8;                   // 8 contiguous windows per thread
        if (j0 < NWIN) {
            float s = 0.f;
            #pragma unroll
            for (int t = 0; t < WIN; ++t) s += sq[j0 + t];
            bv = s; bi = j0;
            #pragma unroll
            for (int jj = 1; jj < 8; ++jj) {
                int j = j0 + jj;
                if (j >= NWIN) break;
                s += sq[j + WIN - 1] - sq[j - 1];
                if (s > bv) { bv = s; bi = j; }   // strict > keeps first index
            }
        }
    }
    rv[tid] = bv;
    ri[tid] = bi;
    __syncthreads();

    #pragma unroll
    for (int off = NTHREADS / 2; off > 0; off >>= 1) {
        if (tid < off) {
            float ov = rv[tid + off];
            int   oi = ri[tid + off];
            float mv = rv[tid];
            if (ov > mv || (ov == mv && oi < ri[tid])) {
                rv[tid] = ov; ri[tid] = oi;
            }
        }
        __syncthreads();
    }
    const int idx = ri[0];

    // ---- Pass 3: gather winning 64x64 slice (coalesced float4) -------------
    const float4* src = (const float4*)(xb + (size_t)idx * DIM);
    float4*       dst = (float4*)(out + (size_t)b * WIN * DIM);
    #pragma unroll
    for (int i = tid; i < WIN * DIM / 4; i += NTHREADS) {
        dst[i] = src[i];
    }
}

extern "C" void kernel_launch(void* const* d_in, const int* in_sizes, int n_in,
                              void* d_out, int out_size, void* d_ws, size_t ws_size,
                              hipStream_t stream) {
    const float* x  = (const float*)d_in[0];
    float*      out = (float*)d_out;
    (void)in_sizes; (void)n_in; (void)out_size; (void)d_ws; (void)ws_size;
    SlidingWindowL2Pooling_kernel<<<dim3(BATCH), dim3(NTHREADS), 0, stream>>>(x, out);
}